// SLICLayer_83648783057276
// MI455X (gfx1250) — compile-verified
//
#include <hip/hip_runtime.h>

// SLIC superpixel k-means on gfx1250 using V_WMMA_F32_16X16X4_F32 for the
// pixel->center distance GEMM (fp32 end to end, matching the jax reference).

typedef float v2f __attribute__((ext_vector_type(2)));
typedef float v8f __attribute__((ext_vector_type(8)));

#define B_IMGS 8
#define H_ 224
#define W_ 224
#define HW (H_ * W_)
#define K_SEG 100
#define KP 112              // centers padded to 7 tiles of 16
#define NT 7                // center tiles of N=16
#define FP 8                // feature dim padded 5 -> 8 (two K=4 WMMAs)
#define CSQ_SLOT 5          // |c|^2 stashed in the zero-padded k5 slot of B
#define N_ITER 10
#define BIGF 3.0e38f
// S = sqrt(H*W/K) = 22.4 exactly; ratio = compactness / S = 10 / 22.4
#define RATIO 0.44642857142857145f

// ---------------------------------------------------------------- init ------
__global__ void slic_init(const float* __restrict__ img,
                          float* __restrict__ centers,
                          float* __restrict__ sums) {
  const int k = threadIdx.x;   // 0..111
  const int b = blockIdx.x;    // image
  if (k >= KP) return;
  float* C = centers + ((size_t)b * KP + k) * FP;
  float* S = sums    + ((size_t)b * KP + k) * FP;
#pragma unroll
  for (int i = 0; i < FP; ++i) S[i] = 0.0f;      // ws is poisoned -> must zero
  if (k < K_SEG) {
    const int iy = k / 10, ix = k % 10;          // g = round(sqrt(100)) = 10
    const int py = (int)floorf((iy + 0.5f) * (float)H_ / 10.0f);
    const int px = (int)floorf((ix + 0.5f) * (float)W_ / 10.0f);
    const float* p = img + ((size_t)b * HW + (size_t)py * W_ + px) * 3;
    const float f0 = py * RATIO, f1 = px * RATIO;
    const float r = p[0], g = p[1], bl = p[2];
    C[0] = f0; C[1] = f1; C[2] = r; C[3] = g; C[4] = bl;
    C[CSQ_SLOT] = f0 * f0 + f1 * f1 + r * r + g * g + bl * bl;
    C[6] = 0.0f; C[7] = 0.0f;
  } else {                                       // padded center: unreachable
    C[0] = C[1] = C[2] = C[3] = C[4] = 0.0f;
    C[CSQ_SLOT] = 1.0e30f;                       // never wins the argmin
    C[6] = 0.0f; C[7] = 0.0f;
  }
}

// ------------------------------------------------------- assign (+accum) ----
// Block = 256 threads = 8 waves; each wave owns 16 pixels vs all 112 centers.
// Per wave: 7 tiles x 2 x v_wmma_f32_16x16x4_f32.
__global__ void slic_assign(const float* __restrict__ img,
                            const float* __restrict__ centers,
                            float* __restrict__ sums,
                            int* __restrict__ labels_out,
                            float* __restrict__ mean_out,
                            int writeOut) {
  __shared__ float sC[KP * FP];    // staged centers (3584 B)
  __shared__ float sAcc[KP * 6];   // block-local segment sums (2688 B)
  __shared__ int   sLbl[8 * 16];   // per-wave labels for redistribution

  const int tid  = threadIdx.x;
  const int b    = blockIdx.y;
  const int lane = tid & 31;
  const int wv   = tid >> 5;
  const size_t imgBase = (size_t)b * HW;

  for (int i = tid; i < KP * FP; i += 256)
    sC[i] = centers[(size_t)b * KP * FP + i];
  if (!writeOut)
    for (int i = tid; i < KP * 6; i += 256) sAcc[i] = 0.0f;
  __syncthreads();

  const int col  = lane & 15;          // pixel-in-wave / B column
  const int half = lane >> 4;          // 0: A rows carry k0,k1 ; 1: k2,k3
  const int pix  = (blockIdx.x * 8 + wv) * 16 + col;   // HW = 392*8*16 exactly
  const int py   = pix / W_;
  const int px   = pix - py * W_;
  const float* ip = img + (imgBase + pix) * 3;

  // A matrix (16x8 f32, fed as two 16x4 operands), ISA 7.12.2 layout.
  v2f a0, a1;
  if (half == 0) {
    a0.x = py * RATIO; a0.y = px * RATIO;    // k0,k1 = scaled y,x
    a1.x = ip[2];      a1.y = 0.0f;          // k4 = blue, k5 = 0 (pads |c|^2)
  } else {
    a0.x = ip[0];      a0.y = ip[1];         // k2,k3 = red, green
    a1.x = 0.0f;       a1.y = 0.0f;          // k6,k7 = 0
  }

  float bestS[8];
  int   bestI[8];
#pragma unroll
  for (int v = 0; v < 8; ++v) { bestS[v] = BIGF; bestI[v] = 0; }

#pragma unroll
  for (int t = 0; t < NT; ++t) {
    const int n  = t * 16 + col;             // this lane's center column
    const int k0 = half * 2;
    const float* cp = &sC[n * FP];
    v2f b0; b0.x = cp[k0];     b0.y = cp[k0 + 1];
    v2f b1; b1.x = cp[k0 + 4]; b1.y = cp[k0 + 5];
    const float csq = cp[CSQ_SLOT];

    v8f acc = {0.f, 0.f, 0.f, 0.f, 0.f, 0.f, 0.f, 0.f};
    acc = __builtin_amdgcn_wmma_f32_16x16x4_f32(false, a0, false, b0,
                                                (short)0, acc, false, false);
    acc = __builtin_amdgcn_wmma_f32_16x16x4_f32(false, a1, false, b1,
                                                (short)0, acc, false, false);
#pragma unroll
    for (int v = 0; v < 8; ++v) {
      const float s = csq - 2.0f * acc[v];   // |c|^2 - 2 f.c  (|f|^2 dropped)
      if (s < bestS[v]) { bestS[v] = s; bestI[v] = n; }
    }
  }

  // argmin across the 16 columns of each half (tie -> lowest center index).
#pragma unroll
  for (int v = 0; v < 8; ++v) {
    float s = bestS[v];
    int   i = bestI[v];
#pragma unroll
    for (int off = 8; off >= 1; off >>= 1) {
      const float os = __shfl_xor(s, off, 16);
      const int   oi = __shfl_xor(i, off, 16);
      if (os < s || (os == s && oi < i)) { s = os; i = oi; }
    }
    if (col == 0) sLbl[wv * 16 + half * 8 + v] = i;  // pixel M = v + 8*half
  }
  __syncthreads();

  const int lab = sLbl[wv * 16 + col];       // label of pixel `col` of my wave
  if (writeOut) {
    if (half == 0) {
      labels_out[imgBase + pix] = lab;
      const float* cc = &sC[lab * FP];
      float* mp = mean_out + (imgBase + pix) * 3;
      mp[0] = cc[2]; mp[1] = cc[3]; mp[2] = cc[4];
    }
  } else {
    // lane holds this pixel's A-slice -> accumulate exactly those features.
    if (half == 0) {
      atomicAdd(&sAcc[lab * 6 + 0], a0.x);   // y*ratio
      atomicAdd(&sAcc[lab * 6 + 1], a0.y);   // x*ratio
      atomicAdd(&sAcc[lab * 6 + 4], a1.x);   // blue
      atomicAdd(&sAcc[lab * 6 + 5], 1.0f);   // count
    } else {
      atomicAdd(&sAcc[lab * 6 + 2], a0.x);   // red
      atomicAdd(&sAcc[lab * 6 + 3], a0.y);   // green
    }
    __syncthreads();
    for (int i = tid; i < KP * 6; i += 256) {
      const float val = sAcc[i];
      if (val != 0.0f) {
        const int k = i / 6, s = i % 6;
        atomicAdd(&sums[((size_t)b * KP + k) * FP + s], val);
      }
    }
  }
}

// -------------------------------------------------------------- update ------
__global__ void slic_update(float* __restrict__ centers,
                            float* __restrict__ sums) {
  const int k = threadIdx.x;
  const int b = blockIdx.x;
  if (k >= KP) return;
  float* C = centers + ((size_t)b * KP + k) * FP;
  float* S = sums    + ((size_t)b * KP + k) * FP;
  const float cnt = S[CSQ_SLOT];             // slot 5 of sums = count
  if (cnt > 0.0f) {                          // else: keep old center (ref)
    const float inv = 1.0f / cnt;            // counts >= 1 here == max(cnt,1)
    const float f0 = S[0] * inv, f1 = S[1] * inv, f2 = S[2] * inv;
    const float f3 = S[3] * inv, f4 = S[4] * inv;
    C[0] = f0; C[1] = f1; C[2] = f2; C[3] = f3; C[4] = f4;
    C[CSQ_SLOT] = f0 * f0 + f1 * f1 + f2 * f2 + f3 * f3 + f4 * f4;
  }
#pragma unroll
  for (int i = 0; i < FP; ++i) S[i] = 0.0f;  // ready for next iteration
}

// -------------------------------------------------------------- launch ------
extern "C" void kernel_launch(void* const* d_in, const int* in_sizes, int n_in,
                              void* d_out, int out_size, void* d_ws,
                              size_t ws_size, hipStream_t stream) {
  const float* img = (const float*)d_in[0];          // [8,224,224,3] f32
  float* centers = (float*)d_ws;                     // [8,112,8]
  float* sums    = centers + (size_t)B_IMGS * KP * FP;
  int*   labels_out = (int*)d_out;                   // [8,224,224] int32 bits
  float* mean_out   = (float*)d_out + (size_t)B_IMGS * HW;  // [8,224,224,3]

  slic_init<<<dim3(B_IMGS), dim3(KP), 0, stream>>>(img, centers, sums);

  const dim3 agrid(HW / 128, B_IMGS);                // 392 x 8 blocks
  for (int it = 0; it < N_ITER; ++it) {
    slic_assign<<<agrid, 256, 0, stream>>>(img, centers, sums,
                                           labels_out, mean_out, 0);
    slic_update<<<dim3(B_IMGS), dim3(KP), 0, stream>>>(centers, sums);
  }
  slic_assign<<<agrid, 256, 0, stream>>>(img, centers, sums,
                                         labels_out, mean_out, 1);
}